// Head_88021059764667
// MI455X (gfx1250) — compile-verified
//
#include <hip/hip_runtime.h>

#define BDIM 2
#define TDIM 512
#define CDIM 512
#define HDIM 64

typedef __attribute__((ext_vector_type(2))) float v2f;
typedef __attribute__((ext_vector_type(8))) float v8f;

// ---------------------------------------------------------------------------
// Kernel 1: carry-forward scans over tokens -> prop_time, prop_pitch (NaN = none yet)
// time: abs_t[t] = cumsum(tok-288 for time tokens)+1 ; prop = rint(abs_t/10) once a
//       time token has been seen (abs_t[last[t]] == abs_t[t] since vals are 0 after).
// pitch: value of most recent note token, shifted by +1 position (clamped at T-1).
// ---------------------------------------------------------------------------
__global__ void scan_kernel(const int* __restrict__ tok,
                            float* __restrict__ prop_time,
                            float* __restrict__ prop_pitch) {
  int b = threadIdx.x;
  if (b >= BDIM) return;
  const int* tp = tok + b * TDIM;
  const float kNaN = __builtin_nanf("");
  float acc = 0.0f;
  bool seen = false;
  float cur_note = kNaN;
  for (int i = 0; i < TDIM; ++i) {
    int v = tp[i];
    if (v >= 288) { acc += (float)(v - 288); seen = true; }
    prop_time[b * TDIM + i] = seen ? rintf((acc + 1.0f) / 10.0f) : kNaN;
    if (v < 256) cur_note = (float)((v >= 128 ? v - 128 : v) + 1);
    if (i >= 1) prop_pitch[b * TDIM + i - 1] = cur_note;  // ff shifted by +1
  }
  prop_pitch[b * TDIM + TDIM - 1] = cur_note;
}

// ---------------------------------------------------------------------------
// Kernel 2: QKV GEMM with V_WMMA_F32_16X16X4_F32.
// One wave per (16-row tile of x) x (q/k/v) x (16-col tile of H).
// A (16x4 f32): lane = m + 16*(k>>1), vgpr = k&1  -> float2 load of (K=2h, 2h+1).
// B (4x16 f32): lane = n + 16*(k>>1), vgpr = k&1  -> two strided loads from W.
// k output is pre-scaled by C^-1/2 so the bias term can be added unscaled.
// ---------------------------------------------------------------------------
__global__ void qkv_kernel(const float* __restrict__ x,
                           const float* __restrict__ Wq,
                           const float* __restrict__ Wk,
                           const float* __restrict__ Wv,
                           float* __restrict__ q,
                           float* __restrict__ kp,
                           float* __restrict__ vp) {
  int rowTile = blockIdx.x;        // 0..63 over B*T/16
  int y = blockIdx.y;              // 0..11
  int mat = y >> 2;                // 0=q 1=k 2=v
  int nt = y & 3;                  // H tile
  const float* W = (mat == 0) ? Wq : (mat == 1) ? Wk : Wv;
  float* out = (mat == 0) ? q : (mat == 1) ? kp : vp;

  int lane = threadIdx.x;
  int m = lane & 15;
  int kh = lane >> 4;              // K half (0/1)
  const float* xrow = x + (rowTile * 16 + m) * CDIM;
  const float* Wcol = W + nt * 16 + m;  // n == lane&15 == m

  v8f c = {};
  for (int kk = 0; kk < CDIM; kk += 4) {
    int k0 = kk + 2 * kh;
    v2f a = *(const v2f*)(xrow + k0);
    v2f bm;
    bm.x = Wcol[k0 * HDIM];
    bm.y = Wcol[(k0 + 1) * HDIM];
    c = __builtin_amdgcn_wmma_f32_16x16x4_f32(false, a, false, bm, (short)0, c,
                                              false, false);
  }
  float scale = (mat == 1) ? 0.04419417382415922f : 1.0f;  // 512^-0.5
  int mbase = (lane < 16) ? 0 : 8;
  for (int r = 0; r < 8; ++r)
    out[(rowTile * 16 + mbase + r) * HDIM + nt * 16 + m] = c[r] * scale;
}

// ---------------------------------------------------------------------------
// Kernel 3: relative-embedding bias sums folded into k', v'.
// k'[b,j,c] += sum_i Ek_pos[clip(j-i)+25] + Ek_time[rt(i,j)] + Ek_pitch[rn(i,j)]
// (same for v' with Ev_*). props staged in LDS; tables stay hot in WGP$/L2.
// NaN distance -> index 0, matching jnp.where(isnan, 0, idx).
// ---------------------------------------------------------------------------
__global__ void bias_kernel(const float* __restrict__ prop_time,
                            const float* __restrict__ prop_pitch,
                            const float* __restrict__ Ek_pos,
                            const float* __restrict__ Ev_pos,
                            const float* __restrict__ Ek_time,
                            const float* __restrict__ Ev_time,
                            const float* __restrict__ Ek_pitch,
                            const float* __restrict__ Ev_pitch,
                            float* __restrict__ kp,
                            float* __restrict__ vp) {
  __shared__ float s_pt[TDIM];
  __shared__ float s_pp[TDIM];
  int b = blockIdx.y;
  int tid = threadIdx.x;  // 256
  for (int i = tid; i < TDIM; i += 256) {
    s_pt[i] = prop_time[b * TDIM + i];
    s_pp[i] = prop_pitch[b * TDIM + i];
  }
  __syncthreads();

  int c = tid & 63;
  int j = blockIdx.x * 4 + (tid >> 6);
  float ptj = s_pt[j];
  float ppj = s_pp[j];
  float kacc = 0.0f, vacc = 0.0f;
  for (int i = 0; i < TDIM; ++i) {
    int dp = j - i;
    dp = dp < -25 ? -25 : (dp > 25 ? 25 : dp);
    int rpos = dp + 25;

    float dt = ptj - s_pt[i];
    int rt = 0;
    if (!(dt != dt)) {
      dt = fminf(200.0f, fmaxf(-200.0f, dt));
      rt = (int)(dt + 200.0f);
    }
    float dn = ppj - s_pp[i];
    int rn = 0;
    if (!(dn != dn)) {
      dn = fminf(128.0f, fmaxf(-128.0f, dn));
      rn = (int)(dn + 128.0f);
    }
    kacc += Ek_pos[rpos * HDIM + c] + Ek_time[rt * HDIM + c] +
            Ek_pitch[rn * HDIM + c];
    vacc += Ev_pos[rpos * HDIM + c] + Ev_time[rt * HDIM + c] +
            Ev_pitch[rn * HDIM + c];
  }
  int idx = (b * TDIM + j) * HDIM + c;
  kp[idx] += kacc;
  vp[idx] += vacc;
}

// ---------------------------------------------------------------------------
// Kernel 4: wei = q @ k'^T  (unmasked; softmax kernel applies causal mask)
// One wave per (b, 16x16 tile). K = H = 64 -> 16 WMMA steps.
// ---------------------------------------------------------------------------
__global__ void wei_kernel(const float* __restrict__ q,
                           const float* __restrict__ kp,
                           float* __restrict__ wei) {
  int jT = blockIdx.x, tT = blockIdx.y, b = blockIdx.z;
  int lane = threadIdx.x;
  int r16 = lane & 15;
  int kh = lane >> 4;
  const float* qrow = q + (b * TDIM + tT * 16 + r16) * HDIM;
  const float* krow = kp + (b * TDIM + jT * 16 + r16) * HDIM;
  v8f c = {};
  for (int kk = 0; kk < HDIM; kk += 4) {
    int k0 = kk + 2 * kh;
    v2f a = *(const v2f*)(qrow + k0);
    v2f bm = *(const v2f*)(krow + k0);
    c = __builtin_amdgcn_wmma_f32_16x16x4_f32(false, a, false, bm, (short)0, c,
                                              false, false);
  }
  int mbase = (lane < 16) ? 0 : 8;
  for (int r = 0; r < 8; ++r)
    wei[(b * TDIM + tT * 16 + mbase + r) * TDIM + jT * 16 + r16] = c[r];
}

// ---------------------------------------------------------------------------
// Kernel 5: causal softmax, one wave32 per row; masked lanes write 0.
// ---------------------------------------------------------------------------
__global__ void softmax_kernel(float* __restrict__ wei) {
  int wid = threadIdx.x >> 5;
  int lane = threadIdx.x & 31;
  int row = blockIdx.x * 8 + wid;  // 0..B*T-1
  int t = row & (TDIM - 1);
  float* w = wei + row * TDIM;

  float vals[16];
  float mx = -__builtin_inff();
  for (int s = 0; s < 16; ++s) {
    int j = lane + 32 * s;
    float v = w[j];
    vals[s] = v;
    if (j <= t && v > mx) mx = v;
  }
  for (int off = 16; off >= 1; off >>= 1) mx = fmaxf(mx, __shfl_xor(mx, off, 32));

  float sum = 0.0f;
  for (int s = 0; s < 16; ++s) {
    int j = lane + 32 * s;
    float e = (j <= t) ? expf(vals[s] - mx) : 0.0f;
    vals[s] = e;
    sum += e;
  }
  for (int off = 16; off >= 1; off >>= 1) sum += __shfl_xor(sum, off, 32);

  float inv = 1.0f / sum;
  for (int s = 0; s < 16; ++s) w[lane + 32 * s] = vals[s] * inv;
}

// ---------------------------------------------------------------------------
// Kernel 6: out = wei @ v'. One wave per (b, tTile, cTile). K = T = 512.
// ---------------------------------------------------------------------------
__global__ void out_kernel(const float* __restrict__ wei,
                           const float* __restrict__ vp,
                           float* __restrict__ out) {
  int cT = blockIdx.x, tT = blockIdx.y, b = blockIdx.z;
  int lane = threadIdx.x;
  int r16 = lane & 15;
  int kh = lane >> 4;
  const float* prow = wei + (b * TDIM + tT * 16 + r16) * TDIM;
  const float* vcol = vp + b * TDIM * HDIM + cT * 16 + r16;
  v8f c = {};
  for (int jj = 0; jj < TDIM; jj += 4) {
    int k0 = jj + 2 * kh;
    v2f a = *(const v2f*)(prow + k0);
    v2f bm;
    bm.x = vcol[k0 * HDIM];
    bm.y = vcol[(k0 + 1) * HDIM];
    c = __builtin_amdgcn_wmma_f32_16x16x4_f32(false, a, false, bm, (short)0, c,
                                              false, false);
  }
  int mbase = (lane < 16) ? 0 : 8;
  for (int r = 0; r < 8; ++r)
    out[(b * TDIM + tT * 16 + mbase + r) * HDIM + cT * 16 + r16] = c[r];
}

extern "C" void kernel_launch(void* const* d_in, const int* in_sizes, int n_in,
                              void* d_out, int out_size, void* d_ws,
                              size_t ws_size, hipStream_t stream) {
  const float* x = (const float*)d_in[0];
  const int* tok = (const int*)d_in[1];
  const float* Wk = (const float*)d_in[2];
  const float* Wq = (const float*)d_in[3];
  const float* Wv = (const float*)d_in[4];
  const float* Ek_pos = (const float*)d_in[5];
  const float* Ev_pos = (const float*)d_in[6];
  const float* Ek_time = (const float*)d_in[7];
  const float* Ev_time = (const float*)d_in[8];
  const float* Ek_pitch = (const float*)d_in[9];
  const float* Ev_pitch = (const float*)d_in[10];
  float* out = (float*)d_out;

  float* ws = (float*)d_ws;
  float* prop_time = ws;                        // B*T
  float* prop_pitch = prop_time + BDIM * TDIM;  // B*T
  float* q = prop_pitch + BDIM * TDIM;          // B*T*H
  float* kp = q + BDIM * TDIM * HDIM;           // B*T*H
  float* vp = kp + BDIM * TDIM * HDIM;          // B*T*H
  float* wei = vp + BDIM * TDIM * HDIM;         // B*T*T

  scan_kernel<<<1, 32, 0, stream>>>(tok, prop_time, prop_pitch);
  qkv_kernel<<<dim3(BDIM * TDIM / 16, 12), 32, 0, stream>>>(x, Wq, Wk, Wv, q,
                                                            kp, vp);
  bias_kernel<<<dim3(TDIM / 4, BDIM), 256, 0, stream>>>(
      prop_time, prop_pitch, Ek_pos, Ev_pos, Ek_time, Ev_time, Ek_pitch,
      Ev_pitch, kp, vp);
  wei_kernel<<<dim3(TDIM / 16, TDIM / 16, BDIM), 32, 0, stream>>>(q, kp, wei);
  softmax_kernel<<<BDIM * TDIM / 8, 256, 0, stream>>>(wei);
  out_kernel<<<dim3(HDIM / 16, TDIM / 16, BDIM), 32, 0, stream>>>(wei, vp, out);
}